// MatrixGATVAE_MT_47425028882912
// MI455X (gfx1250) — compile-verified
//
#include <hip/hip_runtime.h>
#include <hip/hip_bf16.h>

typedef __attribute__((ext_vector_type(2))) float v2f;
typedef __attribute__((ext_vector_type(4))) float v4f;
typedef __attribute__((ext_vector_type(8))) float v8f;

#define NEG_SLOPE 0.2f
#define BN_EPS 1e-5f

// ---------------------------------------------------------------------------
// Order-preserving float<->uint mapping for atomic max on possibly-negative e
// ---------------------------------------------------------------------------
__device__ __forceinline__ unsigned int f2ord(float f) {
  unsigned int u = __float_as_uint(f);
  return (u & 0x80000000u) ? ~u : (u | 0x80000000u);
}
__device__ __forceinline__ float ord2f(unsigned int o) {
  unsigned int u = (o & 0x80000000u) ? (o ^ 0x80000000u) : ~o;
  return __uint_as_float(u);
}

// ---------------------------------------------------------------------------
// Zero-fill (also used for uint buffers; bit pattern of 0.0f == 0u)
// ---------------------------------------------------------------------------
__global__ void zero_u32(unsigned int* __restrict__ p, size_t n) {
  size_t i = (size_t)blockIdx.x * blockDim.x + threadIdx.x;
  if (i < n) p[i] = 0u;
}

// ---------------------------------------------------------------------------
// Conv1d(1, 64, kernel=24, stride=24) on flat = x.transpose(1,2).reshape(N,768)
// h[n, o*32 + t] = conv_b[o] + sum_k x[n, k, t] * conv_w[o, k]
// x is [N, 24, 32] row-major.
// ---------------------------------------------------------------------------
__global__ void conv_kernel(const float* __restrict__ x,
                            const float* __restrict__ w,
                            const float* __restrict__ b,
                            float* __restrict__ h, int Nn) {
  int id = blockIdx.x * 256 + threadIdx.x;
  if (id >= Nn * 2048) return;
  int n   = id >> 11;
  int rem = id & 2047;
  int o   = rem >> 5;
  int t   = rem & 31;
  const float* xp = x + (size_t)n * 768 + t;  // x[n, k, t], stride 32 in k
  const float* wp = w + o * 24;
  float acc = b[o];
#pragma unroll
  for (int k = 0; k < 24; ++k) acc = fmaf(xp[k * 32], wp[k], acc);
  h[id] = acc;
}

// ---------------------------------------------------------------------------
// fp32 WMMA GEMM: C[M,Nc] = A[M,K] @ B[K,Nc] + bias[Nc]
// Block: 256 threads (8 waves). Block tile: 32(M) x 64(N), K-chunk 32.
// Wave grid inside block: 2 (M) x 4 (N); each wave owns one 16x16 tile and
// issues 8x V_WMMA_F32_16X16X4_F32 per K-chunk. The shared LDS A-tile gives
// 4x reuse of the HBM-resident A stream (h is 164MB > L2); B (weights, <=4MB)
// is L2-resident so its re-reads are cheap.
// Staging is fully 128-bit: 1 float4 of A + 2 float4 of B per thread per
// K-chunk (global_load_b128 -> ds_store_b128). As is padded +4 so LDS rows
// stay 16B-aligned (144B stride) AND column reads are bank-conflict-free
// (bank stride 36 mod 64 covers 16 distinct banks since 9 is coprime to 16).
// Fragment packing per ISA 7.12.2 (32-bit A 16x4 / B 4x16 layouts):
//   half = lane>>4 owns K = {2*half, 2*half+1} in vgprs {0,1}.
// Requires: M % 32 == 0, K % 32 == 0, Nc % 64 == 0 (true for all call sites).
// ---------------------------------------------------------------------------
__global__ __launch_bounds__(256)
void wmma_gemm_bias(const float* __restrict__ A, const float* __restrict__ B,
                    const float* __restrict__ bias, float* __restrict__ C,
                    int M, int K, int Nc) {
  __shared__ __align__(16) float As[32][36];  // +4 pad: 16B rows, no bank conflicts
  __shared__ __align__(16) float Bs[32][64];

  const int tid   = threadIdx.x;   // 0..255
  const int wave  = tid >> 5;      // 0..7
  const int waveM = wave & 1;      // 0..1
  const int waveN = wave >> 1;     // 0..3
  const int lane  = tid & 31;
  const int half  = lane >> 4;     // 0: K lo pair, 1: K hi pair
  const int lm    = lane & 15;
  const int rowBase = blockIdx.y * 32 + waveM * 16;
  const int colBase = blockIdx.x * 64 + waveN * 16;

  // Per-thread 128-bit staging assignments (fixed across K loop)
  const int aRow = tid >> 3;        // 0..31
  const int aCol4 = (tid & 7) * 4;  // 0,4,...,28
  const int bRow0 = tid >> 4;       // 0..15  (second half: +16)
  const int bCol4 = (tid & 15) * 4; // 0..60

  const float* Aptr = A + (size_t)(blockIdx.y * 32 + aRow) * K + aCol4;
  const float* Bptr0 = B + (size_t)bRow0 * Nc + blockIdx.x * 64 + bCol4;
  const float* Bptr1 = Bptr0 + (size_t)16 * Nc;

  v8f acc = {};

  for (int kc = 0; kc < K; kc += 32) {
    v4f av = *(const v4f*)(Aptr + kc);
    v4f bv0 = *(const v4f*)(Bptr0 + (size_t)kc * Nc);
    v4f bv1 = *(const v4f*)(Bptr1 + (size_t)kc * Nc);
    *(v4f*)&As[aRow][aCol4] = av;
    *(v4f*)&Bs[bRow0][bCol4] = bv0;
    *(v4f*)&Bs[bRow0 + 16][bCol4] = bv1;
    __syncthreads();

    const int rA = waveM * 16 + lm;
    const int cB = waveN * 16 + lm;
#pragma unroll
    for (int kk = 0; kk < 32; kk += 4) {
      v2f a, b;
      a.x = As[rA][kk + 2 * half];
      a.y = As[rA][kk + 2 * half + 1];
      b.x = Bs[kk + 2 * half][cB];
      b.y = Bs[kk + 2 * half + 1][cB];
      acc = __builtin_amdgcn_wmma_f32_16x16x4_f32(
          /*neg_a=*/false, a, /*neg_b=*/false, b,
          /*c_mod=*/(short)0, acc, /*reuse_a=*/false, /*reuse_b=*/false);
    }
    __syncthreads();
  }

  // C/D layout: acc[r] -> row = r + 8*half, col = lm
  float bvv = bias[colBase + lm];
#pragma unroll
  for (int r = 0; r < 8; ++r) {
    int row = rowBase + half * 8 + r;
    C[(size_t)row * Nc + colBase + lm] = acc[r] + bvv;
  }
}

// ---------------------------------------------------------------------------
// Edge pass 1: e = att . leaky_relu(xl[src] + xr[dst]); per-dst atomic max.
// One wave (32 lanes) per edge. Self-loops are the last Nn "virtual" edges.
// ---------------------------------------------------------------------------
__global__ __launch_bounds__(256)
void edge_e_max(const float* __restrict__ xl, const float* __restrict__ xr,
                const float* __restrict__ att,
                const int* __restrict__ src, const int* __restrict__ dst,
                int E0, int Nn, int D,
                float* __restrict__ e, unsigned int* __restrict__ m) {
  int edge = blockIdx.x * 8 + (threadIdx.x >> 5);
  int lane = threadIdx.x & 31;
  if (edge >= E0 + Nn) return;
  int s = (edge < E0) ? src[edge] : (edge - E0);
  int d = (edge < E0) ? dst[edge] : (edge - E0);
  float partial = 0.f;
  for (int j = lane; j < D; j += 32) {
    float v = xl[(size_t)s * D + j] + xr[(size_t)d * D + j];
    v = (v > 0.f) ? v : NEG_SLOPE * v;
    partial = fmaf(v, att[j], partial);
  }
#pragma unroll
  for (int off = 16; off > 0; off >>= 1)
    partial += __shfl_xor(partial, off, 32);
  if (lane == 0) {
    e[edge] = partial;
    atomicMax(m + d, f2ord(partial));
  }
}

// ---------------------------------------------------------------------------
// Edge pass 2: a = exp(e - m[dst]); s[dst] += a  (a written back over e)
// ---------------------------------------------------------------------------
__global__ void edge_exp_sum(const int* __restrict__ dst, int E0, int Nn,
                             float* __restrict__ e,
                             const unsigned int* __restrict__ m,
                             float* __restrict__ s) {
  int i = blockIdx.x * 256 + threadIdx.x;
  if (i >= E0 + Nn) return;
  int d = (i < E0) ? dst[i] : (i - E0);
  float a = expf(e[i] - ord2f(m[d]));
  e[i] = a;
  atomicAdd(s + d, a);
}

// ---------------------------------------------------------------------------
// Edge pass 3: agg[dst] += xl[src] * (a / (s[dst] + 1e-16)). Wave per edge.
// ---------------------------------------------------------------------------
__global__ __launch_bounds__(256)
void edge_aggregate(const float* __restrict__ xl, const float* __restrict__ a,
                    const float* __restrict__ s,
                    const int* __restrict__ src, const int* __restrict__ dst,
                    int E0, int Nn, int D, float* __restrict__ agg) {
  int edge = blockIdx.x * 8 + (threadIdx.x >> 5);
  int lane = threadIdx.x & 31;
  if (edge >= E0 + Nn) return;
  int sN = (edge < E0) ? src[edge] : (edge - E0);
  int dN = (edge < E0) ? dst[edge] : (edge - E0);
  float alpha = a[edge] / (s[dN] + 1e-16f);
  for (int j = lane; j < D; j += 32)
    atomicAdd(agg + (size_t)dN * D + j, xl[(size_t)sN * D + j] * alpha);
}

// ---------------------------------------------------------------------------
// y = relu(agg + bias); write back; column partial sums for batchnorm stats.
// Thread owns one column for a chunk of rows -> few atomics per counter.
// ---------------------------------------------------------------------------
__global__ void post_relu_stats(float* __restrict__ h,
                                const float* __restrict__ bias,
                                float* __restrict__ sum, float* __restrict__ ssq,
                                int Nn, int D, int rowsPerBlock) {
  int c = blockIdx.x * blockDim.x + threadIdx.x;
  if (c >= D) return;
  int r0 = blockIdx.y * rowsPerBlock;
  int r1 = r0 + rowsPerBlock; if (r1 > Nn) r1 = Nn;
  float bv = bias[c], ls = 0.f, lq = 0.f;
  for (int r = r0; r < r1; ++r) {
    float y = h[(size_t)r * D + c] + bv;
    y = (y > 0.f) ? y : 0.f;
    h[(size_t)r * D + c] = y;
    ls += y;
    lq = fmaf(y, y, lq);
  }
  atomicAdd(sum + c, ls);
  atomicAdd(ssq + c, lq);
}

__global__ void bn_apply(float* __restrict__ h, const float* __restrict__ sum,
                         const float* __restrict__ ssq,
                         const float* __restrict__ gamma,
                         const float* __restrict__ beta, int Nn, int D) {
  size_t i = (size_t)blockIdx.x * 256 + threadIdx.x;
  if (i >= (size_t)Nn * D) return;
  int c = (int)(i % D);
  float invN = 1.f / (float)Nn;
  float mean = sum[c] * invN;
  float var  = fmaxf(ssq[c] * invN - mean * mean, 0.f);
  float inv  = rsqrtf(var + BN_EPS);
  h[i] = gamma[c] * (h[i] - mean) * inv + beta[c];
}

__global__ void add_bias(float* __restrict__ out, const float* __restrict__ bias,
                         int Nn, int D) {
  int i = blockIdx.x * 256 + threadIdx.x;
  if (i < Nn * D) out[i] += bias[i % D];
}

// ---------------------------------------------------------------------------
extern "C" void kernel_launch(void* const* d_in, const int* in_sizes, int n_in,
                              void* d_out, int out_size, void* d_ws, size_t ws_size,
                              hipStream_t stream) {
  const float* x      = (const float*)d_in[0];
  const int*   ei     = (const int*)  d_in[1];
  const float* conv_w = (const float*)d_in[2];
  const float* conv_b = (const float*)d_in[3];
  const float* W1l    = (const float*)d_in[4];
  const float* b1l    = (const float*)d_in[5];
  const float* W1r    = (const float*)d_in[6];
  const float* b1r    = (const float*)d_in[7];
  const float* att1   = (const float*)d_in[8];
  const float* bias1  = (const float*)d_in[9];
  const float* gamma  = (const float*)d_in[10];
  const float* beta   = (const float*)d_in[11];
  const float* W2l    = (const float*)d_in[12];
  const float* b2l    = (const float*)d_in[13];
  const float* W2r    = (const float*)d_in[14];
  const float* b2r    = (const float*)d_in[15];
  const float* att2   = (const float*)d_in[16];
  const float* bias2  = (const float*)d_in[17];

  const int Nn = in_sizes[0] / (24 * 32);   // 20000
  const int E0 = in_sizes[1] / 2;           // 320000
  const int E  = E0 + Nn;                   // + self loops
  const int D1 = 512, D2 = 128, IN = 2048;
  const int* srcIdx = ei;
  const int* dstIdx = ei + E0;

  // Workspace layout (fp32 elements)
  float* ws  = (float*)d_ws;
  float* h   = ws;                               // [Nn,2048]; reused as agg1
  float* xl1 = h   + (size_t)Nn * IN;            // [Nn,512]
  float* xr1 = xl1 + (size_t)Nn * D1;            // [Nn,512]
  float* xl2 = xr1 + (size_t)Nn * D1;            // [Nn,128]
  float* xr2 = xl2 + (size_t)Nn * D2;            // [Nn,128]
  float* ebuf = xr2 + (size_t)Nn * D2;           // [E]
  unsigned int* mbuf = (unsigned int*)(ebuf + E);// [Nn]
  float* sbuf = (float*)(mbuf + Nn);             // [Nn]
  float* sums = sbuf + Nn;                       // [512]
  float* ssqs = sums + D1;                       // [512]
  float* agg1 = h;                               // alias: h dead after GEMM1
  float* outp = (float*)d_out;                   // [Nn,128]

  const int edgeBlocks = (E + 7) / 8;

  // 1) Conv1d -> h
  conv_kernel<<<(Nn * 2048 + 255) / 256, 256, 0, stream>>>(x, conv_w, conv_b, h, Nn);

  // 2) Layer-1 projections (fp32 WMMA GEMMs)
  {
    dim3 grid(D1 / 64, Nn / 32);
    wmma_gemm_bias<<<grid, 256, 0, stream>>>(h, W1l, b1l, xl1, Nn, IN, D1);
    wmma_gemm_bias<<<grid, 256, 0, stream>>>(h, W1r, b1r, xr1, Nn, IN, D1);
  }

  // 3) Layer-1 attention + aggregation (agg1 aliases h; GEMMs done reading it)
  zero_u32<<<(Nn + 255) / 256, 256, 0, stream>>>(mbuf, (size_t)Nn);
  zero_u32<<<(Nn + 255) / 256, 256, 0, stream>>>((unsigned int*)sbuf, (size_t)Nn);
  zero_u32<<<((size_t)Nn * D1 + 255) / 256, 256, 0, stream>>>((unsigned int*)agg1, (size_t)Nn * D1);
  edge_e_max<<<edgeBlocks, 256, 0, stream>>>(xl1, xr1, att1, srcIdx, dstIdx, E0, Nn, D1, ebuf, mbuf);
  edge_exp_sum<<<(E + 255) / 256, 256, 0, stream>>>(dstIdx, E0, Nn, ebuf, mbuf, sbuf);
  edge_aggregate<<<edgeBlocks, 256, 0, stream>>>(xl1, ebuf, sbuf, srcIdx, dstIdx, E0, Nn, D1, agg1);

  // 4) bias + ReLU + batch-stats, then BN normalize (in place on agg1)
  zero_u32<<<1, 2 * D1, 0, stream>>>((unsigned int*)sums, (size_t)(2 * D1));
  {
    dim3 grid(D1 / 256, (Nn + 255) / 256);
    post_relu_stats<<<grid, 256, 0, stream>>>(agg1, bias1, sums, ssqs, Nn, D1, 256);
  }
  bn_apply<<<((size_t)Nn * D1 + 255) / 256, 256, 0, stream>>>(agg1, sums, ssqs, gamma, beta, Nn, D1);

  // 5) Layer-2 projections
  {
    dim3 grid(D2 / 64, Nn / 32);
    wmma_gemm_bias<<<grid, 256, 0, stream>>>(agg1, W2l, b2l, xl2, Nn, D1, D2);
    wmma_gemm_bias<<<grid, 256, 0, stream>>>(agg1, W2r, b2r, xr2, Nn, D1, D2);
  }

  // 6) Layer-2 attention + aggregation -> d_out, + bias2
  zero_u32<<<(Nn + 255) / 256, 256, 0, stream>>>(mbuf, (size_t)Nn);
  zero_u32<<<(Nn + 255) / 256, 256, 0, stream>>>((unsigned int*)sbuf, (size_t)Nn);
  zero_u32<<<((size_t)Nn * D2 + 255) / 256, 256, 0, stream>>>((unsigned int*)outp, (size_t)Nn * D2);
  edge_e_max<<<edgeBlocks, 256, 0, stream>>>(xl2, xr2, att2, srcIdx, dstIdx, E0, Nn, D2, ebuf, mbuf);
  edge_exp_sum<<<(E + 255) / 256, 256, 0, stream>>>(dstIdx, E0, Nn, ebuf, mbuf, sbuf);
  edge_aggregate<<<edgeBlocks, 256, 0, stream>>>(xl2, ebuf, sbuf, srcIdx, dstIdx, E0, Nn, D2, outp);
  add_bias<<<(Nn * D2 + 255) / 256, 256, 0, stream>>>(outp, bias2, Nn, D2);
}